// AttentionAggregation_12790412607648
// MI455X (gfx1250) — compile-verified
//
#include <hip/hip_runtime.h>
#include <hip/hip_bf16.h>
#include <math.h>

typedef float v2f __attribute__((ext_vector_type(2)));
typedef float v4f __attribute__((ext_vector_type(4)));
typedef float v8f __attribute__((ext_vector_type(8)));

#define D 128
#define NSEG 4096

// LDS layout (in floats)
#define W1T_STRIDE 132                      // frag reads: banks 4n (lanes) / +2 (halves) -> conflict-free
#define XS_STRIDE  36                       // 144B rows: 16B-aligned, banks 4*(9r%16) -> conflict-free
#define W1T_OFF 0
#define XS_OFF  (128 * W1T_STRIDE)          // 16896
#define B1_OFF  (XS_OFF + 128 * XS_STRIDE)  // 21504
#define W2_OFF  (B1_OFF + 128)              // 21632
#define LDS_FLOATS (W2_OFF + 128)           // 21760 floats = 87040 B

// ---------------------------------------------------------------------------
// Pass 1: scores[i] = tanh(x[i] @ W1 + b1) @ W2 + b2 via V_WMMA_F32_16X16X4_F32.
// Persistent grid-stride blocks; W1 staged transposed once per block.
// Software-pipelined double buffer enforced with sched_barrier(0); occupancy
// is LDS-bound (3 blocks/WGP), so launch_bounds(256,1) lifts the VGPR cap to
// hold ~170 live VGPRs (64 acc + 16 A + 64 B double buffer) without spilling.
// ---------------------------------------------------------------------------
__global__ void __launch_bounds__(256, 1)
scores_kernel(const float* __restrict__ x,
              const float* __restrict__ W1,
              const float* __restrict__ b1,
              const float* __restrict__ W2,
              const float* __restrict__ b2,
              float* __restrict__ scores, int n, int numTiles)
{
    extern __shared__ float lds[];
    const int tid = threadIdx.x;

    // Stage W1 transposed once: W1 is [K=128][N=128] row-major -> lds[n][k].
    for (int idx4 = tid; idx4 < (D * D) / 4; idx4 += 256) {
        int k  = idx4 >> 5;            // 32 float4 per K-row
        int n4 = (idx4 & 31) * 4;
        v4f w = ((const v4f*)W1)[idx4];
        lds[W1T_OFF + (n4 + 0) * W1T_STRIDE + k] = w.x;
        lds[W1T_OFF + (n4 + 1) * W1T_STRIDE + k] = w.y;
        lds[W1T_OFF + (n4 + 2) * W1T_STRIDE + k] = w.z;
        lds[W1T_OFF + (n4 + 3) * W1T_STRIDE + k] = w.w;
    }
    if (tid < D) {
        lds[B1_OFF + tid] = b1[tid];
        lds[W2_OFF + tid] = W2[tid];
    }
    // (first __syncthreads inside the chunk loop publishes W1T before any use)

    const int wave = tid >> 5;
    const int lane = tid & 31;
    const int hi   = lane >> 4;   // half-wave: selects K pair (A) / K rows (B)
    const int nIdx = lane & 15;
    const float b2v = b2[0];

    for (int tile = blockIdx.x; tile < numTiles; tile += gridDim.x) {
        const int blockRow = tile * 128;
        v8f acc[8] = {};   // 8 column tiles of the 16x128 hidden pre-activation

        for (int chunk = 0; chunk < 4; ++chunk) {
            __syncthreads();  // previous readers of Xs (and W1T stores) done
            // Stage X chunk: 128 rows x 32 K, b128 both sides.
            for (int idx4 = tid; idx4 < (128 * 32) / 4; idx4 += 256) {
                int r  = idx4 >> 3;            // 8 float4 per row-chunk
                int c4 = (idx4 & 7) * 4;
                int grow = blockRow + r;
                v4f v = {0.0f, 0.0f, 0.0f, 0.0f};
                if (grow < n)
                    v = *(const v4f*)&x[(size_t)grow * D + chunk * 32 + c4];
                *(v4f*)&lds[XS_OFF + r * XS_STRIDE + c4] = v;
            }
            __syncthreads();

            // A fragments (16 rows of this wave): lane<16 holds K={k0,k0+1},
            // lane>=16 K={k0+2,k0+3}, row = nIdx.
            const float* ap = &lds[XS_OFF + (16 * wave + nIdx) * XS_STRIDE + 2 * hi];
            v2f a8[8];
#pragma unroll
            for (int kk = 0; kk < 8; ++kk)
                a8[kk] = *(const v2f*)(ap + kk * 4);

            const float* bbase = &lds[W1T_OFF + nIdx * W1T_STRIDE + chunk * 32 + 2 * hi];

            // Quarter q covers K-steps kk = 2q, 2q+1 (8 loads, 16 WMMAs).
            auto loadQ = [&](v2f (&buf)[8][2], int q) {
#pragma unroll
                for (int t = 0; t < 8; ++t) {
                    const float* bp = bbase + (16 * t) * W1T_STRIDE + q * 8;
                    buf[t][0] = *(const v2f*)(bp);
                    buf[t][1] = *(const v2f*)(bp + 4);
                }
            };
            auto compQ = [&](v2f (&buf)[8][2], int q) {
#pragma unroll
                for (int kk = 0; kk < 2; ++kk) {
#pragma unroll
                    for (int t = 0; t < 8; ++t)
                        acc[t] = __builtin_amdgcn_wmma_f32_16x16x4_f32(
                            false, a8[q * 2 + kk], false, buf[t][kk],
                            (short)0, acc[t], false, false);
                }
            };

            v2f bufA[8][2], bufB[8][2];
            // Software pipeline: L0 L1 | C0 L2 | C1 L3 | C2 | C3
            loadQ(bufA, 0);
            __builtin_amdgcn_sched_barrier(0);
            loadQ(bufB, 1);
            __builtin_amdgcn_sched_barrier(0);
            compQ(bufA, 0);
            __builtin_amdgcn_sched_barrier(0);
            loadQ(bufA, 2);
            __builtin_amdgcn_sched_barrier(0);
            compQ(bufB, 1);
            __builtin_amdgcn_sched_barrier(0);
            loadQ(bufB, 3);
            __builtin_amdgcn_sched_barrier(0);
            compQ(bufA, 2);
            __builtin_amdgcn_sched_barrier(0);
            compQ(bufB, 3);
        }

        // Finalize: C/D layout -> VGPR g holds row (g + 8*hi), col = 16t + nIdx.
        float p[8];
#pragma unroll
        for (int g = 0; g < 8; ++g) p[g] = 0.0f;
#pragma unroll
        for (int t = 0; t < 8; ++t) {
            const int ncol = 16 * t + nIdx;
            const float bb = lds[B1_OFF + ncol];
            const float ww = lds[W2_OFF + ncol];
#pragma unroll
            for (int g = 0; g < 8; ++g)
                p[g] += tanhf(acc[t][g] + bb) * ww;
        }
        // Reduce across the 16 lanes of each half-wave (wave32 shuffles).
#pragma unroll
        for (int m = 1; m < 16; m <<= 1) {
#pragma unroll
            for (int g = 0; g < 8; ++g)
                p[g] += __shfl_xor(p[g], m, 32);
        }
        if (nIdx == 0) {  // lane 0 -> rows 0..7, lane 16 -> rows 8..15
            const int base = blockRow + 16 * wave + 8 * hi;
#pragma unroll
            for (int g = 0; g < 8; ++g) {
                int r = base + g;
                if (r < n) scores[r] = p[g] + b2v;
            }
        }
    }
}

// ---------------------------------------------------------------------------
// Pass 2: segment boundaries via binary search on sorted batch_idx.
// seg_start[s] = first i with idx[i] >= s ; seg_start[NSEG] = n.
// ---------------------------------------------------------------------------
__global__ void seg_bounds(const int* __restrict__ idx, int n, int nseg,
                           int* __restrict__ seg_start)
{
    int s = blockIdx.x * blockDim.x + threadIdx.x;
    if (s > nseg) return;
    if (s == nseg) { seg_start[s] = n; return; }
    int lo = 0, hi = n;
    while (lo < hi) {
        int mid = (lo + hi) >> 1;
        if (idx[mid] < s) lo = mid + 1; else hi = mid;
    }
    seg_start[s] = lo;
}

// ---------------------------------------------------------------------------
// Pass 3: per-segment softmax + alpha-weighted aggregation of x.
// One block (256 threads) per segment; e written back over scores scratch.
// ---------------------------------------------------------------------------
__global__ void seg_softmax_agg(const float* __restrict__ x,
                                float* __restrict__ e_scores,
                                const int* __restrict__ seg_start,
                                float* __restrict__ out,
                                float* __restrict__ alpha, int n)
{
    __shared__ float red[256];
    const int seg = blockIdx.x;
    const int tid = threadIdx.x;
    const int beg = seg_start[seg];
    const int end = seg_start[seg + 1];

    // max
    float m = -__builtin_inff();
    for (int i = beg + tid; i < end; i += 256) m = fmaxf(m, e_scores[i]);
    red[tid] = m; __syncthreads();
    for (int s = 128; s > 0; s >>= 1) {
        if (tid < s) red[tid] = fmaxf(red[tid], red[tid + s]);
        __syncthreads();
    }
    const float smax = (end > beg) ? red[0] : 0.0f;  // isfinite(seg_max) ? . : 0
    __syncthreads();

    // exp + sum (store e in place)
    float loc = 0.0f;
    for (int i = beg + tid; i < end; i += 256) {
        float e = expf(e_scores[i] - smax);
        e_scores[i] = e;
        loc += e;
    }
    red[tid] = loc; __syncthreads();
    for (int s = 128; s > 0; s >>= 1) {
        if (tid < s) red[tid] += red[tid + s];
        __syncthreads();
    }
    const float inv = 1.0f / (red[0] + 1e-16f);
    __syncthreads();

    // alpha
    for (int i = beg + tid; i < end; i += 256) alpha[i] = e_scores[i] * inv;

    // weighted column sums: 2 row-groups x 128 cols (coalesced x reads)
    const int c = tid & 127;
    const int g = tid >> 7;
    float a = 0.0f;
    for (int i = beg + g; i < end; i += 2)
        a += x[(size_t)i * D + c] * (e_scores[i] * inv);
    red[tid] = a; __syncthreads();
    if (tid < 128)
        out[(size_t)seg * D + c] = red[tid] + red[tid + 128];
}

// ---------------------------------------------------------------------------
extern "C" void kernel_launch(void* const* d_in, const int* in_sizes, int n_in,
                              void* d_out, int out_size, void* d_ws, size_t ws_size,
                              hipStream_t stream)
{
    const float* x   = (const float*)d_in[0];
    const float* W1  = (const float*)d_in[1];
    const float* b1  = (const float*)d_in[2];
    const float* W2  = (const float*)d_in[3];
    const float* b2  = (const float*)d_in[4];
    const int*   bi  = (const int*)d_in[5];   // jax default x64-off -> int32
    const int    n   = in_sizes[5];           // N rows

    float* out   = (float*)d_out;
    float* alpha = out + (size_t)NSEG * D;

    float* scores    = (float*)d_ws;          // N floats (then reused for e)
    int*   seg_start = (int*)(scores + n);    // NSEG+1 ints

    const size_t ldsBytes = (size_t)LDS_FLOATS * sizeof(float);
    const int numTiles = (n + 127) / 128;
    const int blocksA  = numTiles < 2048 ? numTiles : 2048;

    scores_kernel<<<blocksA, 256, ldsBytes, stream>>>(x, W1, b1, W2, b2,
                                                      scores, n, numTiles);
    seg_bounds<<<(NSEG + 1 + 255) / 256, 256, 0, stream>>>(bi, n, NSEG, seg_start);
    seg_softmax_agg<<<NSEG, 256, 0, stream>>>(x, scores, seg_start, out, alpha, n);
}